// DGCNNSpatialBranch_5274219839629
// MI455X (gfx1250) — compile-verified
//
#include <hip/hip_runtime.h>
#include <hip/hip_bf16.h>

typedef __attribute__((ext_vector_type(2))) float v2f;
typedef __attribute__((ext_vector_type(8))) float v8f;

#define NBATCH 8
#define NPTS   2048
#define KK     20
#define EPS_BN 1e-5f
#define NEG_SLOPE 0.2f
#define ENC_NEG_INF 0x007FFFFFu

// ---- order-preserving float<->uint encoding for atomicMax over floats ----
__device__ __forceinline__ unsigned enc_f32(float f) {
  unsigned u = __float_as_uint(f);
  return (u & 0x80000000u) ? ~u : (u | 0x80000000u);
}
__device__ __forceinline__ float dec_f32(unsigned u) {
  return __uint_as_float((u & 0x80000000u) ? (u & 0x7FFFFFFFu) : ~u);
}

// ---- 1. transpose (B,C,N) -> padded (B,N,Cp) ----
__global__ void k_transpose_pad(const float* __restrict__ x, float* __restrict__ xp,
                                int C, int Cp) {
  int t = blockIdx.x * blockDim.x + threadIdx.x;
  int total = NBATCH * NPTS * Cp;
  if (t >= total) return;
  int c = t % Cp;
  int n = (t / Cp) % NPTS;
  int b = t / (Cp * NPTS);
  xp[t] = (c < C) ? x[((size_t)b * C + c) * NPTS + n] : 0.0f;
}

// ---- 2. per-point squared norms ----
__global__ void k_row_norms(const float* __restrict__ X, float* __restrict__ sq, int Cp) {
  int t = blockIdx.x * blockDim.x + threadIdx.x;
  if (t >= NBATCH * NPTS) return;
  const float* r = X + (size_t)t * Cp;
  float s = 0.0f;
  for (int c = 0; c < Cp; ++c) s += r[c] * r[c];
  sq[t] = s;
}

// ---- 3. negative pairwise distances for one batch via V_WMMA_F32_16X16X4_F32 ----
// neg_d[i][j] = 2*dot(xi,xj) - |xi|^2 - |xj|^2
__global__ void k_dist_wmma(const float* __restrict__ X, const float* __restrict__ sq,
                            float* __restrict__ dist, int Cp, int batch) {
  const int TI = NPTS / 16;                       // 128 tiles per dim
  int wave = (int)((blockIdx.x * blockDim.x + threadIdx.x) >> 5);
  int lane = threadIdx.x & 31;
  if (wave >= TI * TI) return;                    // wave-uniform; EXEC stays all-1s
  int ti = wave / TI, tj = wave % TI;

  const float* Xb  = X  + (size_t)batch * NPTS * Cp;
  const float* sqb = sq + (size_t)batch * NPTS;

  int mrow = ti * 16 + (lane & 15);               // A fragment row (16x4 layout)
  int ncol = tj * 16 + (lane & 15);               // B fragment col (4x16 layout)
  int koff = (lane < 16) ? 0 : 2;                 // half-wave K split

  v8f acc = {};
  for (int k0 = 0; k0 < Cp; k0 += 4) {
    v2f a, bm;
    a.x  = Xb[(size_t)mrow * Cp + k0 + koff];
    a.y  = Xb[(size_t)mrow * Cp + k0 + koff + 1];
    bm.x = Xb[(size_t)ncol * Cp + k0 + koff];
    bm.y = Xb[(size_t)ncol * Cp + k0 + koff + 1];
    acc = __builtin_amdgcn_wmma_f32_16x16x4_f32(false, a, false, bm,
                                                (short)0, acc, false, false);
  }

  float sj = sqb[ncol];
#pragma unroll
  for (int i = 0; i < 8; ++i) {                   // C/D layout: VGPR i -> M=i / M=i+8
    int row = ti * 16 + i + ((lane < 16) ? 0 : 8);
    int col = tj * 16 + (lane & 15);
    dist[(size_t)row * NPTS + col] = 2.0f * acc[i] - sqb[row] - sj;
  }
}

// ---- 4. top-k (k=20) per row via iterative block argmax in LDS ----
__global__ void k_topk(const float* __restrict__ dist, int* __restrict__ idx, int batch) {
  __shared__ float row[NPTS];
  __shared__ float sval[256];
  __shared__ int   sidx[256];
  int n = blockIdx.x;
  const float* d = dist + (size_t)n * NPTS;
  for (int j = threadIdx.x; j < NPTS; j += 256) row[j] = d[j];
  __syncthreads();
  int* out = idx + ((size_t)batch * NPTS + n) * KK;
  for (int sel = 0; sel < KK; ++sel) {
    float best = -__builtin_inff();
    int bi = 0;
    for (int j = threadIdx.x; j < NPTS; j += 256) {
      float v = row[j];
      if (v > best) { best = v; bi = j; }
    }
    sval[threadIdx.x] = best;
    sidx[threadIdx.x] = bi;
    __syncthreads();
    for (int s = 128; s > 0; s >>= 1) {
      if (threadIdx.x < s) {
        if (sval[threadIdx.x + s] > sval[threadIdx.x]) {
          sval[threadIdx.x] = sval[threadIdx.x + s];
          sidx[threadIdx.x] = sidx[threadIdx.x + s];
        }
      }
      __syncthreads();
    }
    if (threadIdx.x == 0) {
      out[sel] = sidx[0];
      row[sidx[0]] = -__builtin_inff();
    }
    __syncthreads();
  }
}

// ---- 5. init BN accumulators + running max buffer ----
__global__ void k_init_stats(float* __restrict__ sums, float* __restrict__ sumsq,
                             unsigned* __restrict__ maxbuf, int O, int total) {
  int t = blockIdx.x * blockDim.x + threadIdx.x;
  if (t < O) { sums[t] = 0.0f; sumsq[t] = 0.0f; }
  if (t < total) maxbuf[t] = ENC_NEG_INF;
}

// ---- 6. edge-feature GEMM h = e * W^T via V_WMMA_F32_16X16X4_F32, fused with
//          per-channel sum/sumsq (BN stats) and max-over-k accumulation ----
__global__ void k_edge_gemm_wmma(const float* __restrict__ X, const int* __restrict__ idx,
                                 const float* __restrict__ W,
                                 float* __restrict__ sums, float* __restrict__ sumsq,
                                 unsigned* __restrict__ maxbuf,
                                 int C, int Cp, int O) {
  const int twoC   = 2 * C;
  const int ntiles = O / 16;
  const int mtiles = (NBATCH * NPTS * KK) / 16;   // 20480
  int wave = (int)((blockIdx.x * blockDim.x + threadIdx.x) >> 5);
  int lane = threadIdx.x & 31;
  if (wave >= mtiles * ntiles) return;            // wave-uniform
  int mt = wave / ntiles, nt = wave % ntiles;

  // This lane's A-fragment edge row (rows enumerate (b, n, kk), kk fastest)
  int r  = mt * 16 + (lane & 15);
  int kk = r % KK;
  int n  = (r / KK) % NPTS;
  int b  = r / (KK * NPTS);
  (void)kk;
  const float* ctr = X + ((size_t)b * NPTS + n) * Cp;
  int nbr = idx[r];
  const float* ftr = X + ((size_t)b * NPTS + nbr) * Cp;

  // This lane's B-fragment output channel
  int oc = nt * 16 + (lane & 15);
  const float* wr = W + (size_t)oc * twoC;

  int koff = (lane < 16) ? 0 : 2;
  v8f acc = {};
  for (int k0 = 0; k0 < twoC; k0 += 4) {
    v2f a, bm;
#pragma unroll
    for (int t2 = 0; t2 < 2; ++t2) {
      int k = k0 + koff + t2;
      float e;
      if (k < C)          e = ftr[k] - ctr[k];    // feat - ctr
      else if (k < twoC)  e = ctr[k - C];         // broadcast ctr
      else                e = 0.0f;               // K padding (layer 1: 2C=10)
      float wv = (k < twoC) ? wr[k] : 0.0f;
      if (t2 == 0) { a.x = e; bm.x = wv; } else { a.y = e; bm.y = wv; }
    }
    acc = __builtin_amdgcn_wmma_f32_16x16x4_f32(false, a, false, bm,
                                                (short)0, acc, false, false);
  }

  // BN statistics + max over k (monotone transform -> max commutes)
  float ls = 0.0f, lq = 0.0f;
#pragma unroll
  for (int i = 0; i < 8; ++i) {
    float v = acc[i];
    ls += v;
    lq += v * v;
    int row = mt * 16 + i + ((lane < 16) ? 0 : 8);
    int nn  = (row / KK) % NPTS;
    int bb  = row / (KK * NPTS);
    atomicMax(&maxbuf[((size_t)bb * NPTS + nn) * O + oc], enc_f32(v));
  }
  atomicAdd(&sums[oc],  ls);
  atomicAdd(&sumsq[oc], lq);
}

// ---- 7. finalize: BN(train) + LeakyReLU applied to the per-(b,n,oc) max ----
__global__ void k_finalize(const unsigned* __restrict__ maxbuf,
                           const float* __restrict__ sums, const float* __restrict__ sumsq,
                           const float* __restrict__ gamma, const float* __restrict__ beta,
                           float* __restrict__ out, int O, int transposed) {
  int t = blockIdx.x * blockDim.x + threadIdx.x;
  int total = NBATCH * NPTS * O;
  if (t >= total) return;
  int oc = t % O;
  int n  = (t / O) % NPTS;
  int b  = t / (O * NPTS);
  const float cnt = (float)(NBATCH * NPTS * KK);
  float mu  = sums[oc] / cnt;
  float var = sumsq[oc] / cnt - mu * mu;
  float v = dec_f32(maxbuf[t]);
  float y = (v - mu) * rsqrtf(var + EPS_BN) * gamma[oc] + beta[oc];
  y = (y > 0.0f) ? y : NEG_SLOPE * y;
  if (transposed) out[((size_t)b * O + oc) * NPTS + n] = y;  // final (B,O,N)
  else            out[t] = y;                                // (B,N,O), stride O == next Cp
}

extern "C" void kernel_launch(void* const* d_in, const int* in_sizes, int n_in,
                              void* d_out, int out_size, void* d_ws, size_t ws_size,
                              hipStream_t stream) {
  (void)in_sizes; (void)n_in; (void)out_size; (void)ws_size;
  const float* x = (const float*)d_in[0];
  const float* Wl[4] = {(const float*)d_in[1], (const float*)d_in[4],
                        (const float*)d_in[7], (const float*)d_in[10]};
  const float* gl[4] = {(const float*)d_in[2], (const float*)d_in[5],
                        (const float*)d_in[8], (const float*)d_in[11]};
  const float* bl[4] = {(const float*)d_in[3], (const float*)d_in[6],
                        (const float*)d_in[9], (const float*)d_in[12]};
  const int Cs[4]  = {5, 64, 64, 128};
  const int Cps[4] = {8, 64, 64, 128};
  const int Os[4]  = {64, 64, 128, 128};

  // ---- workspace carve (bytes) ----
  char* ws = (char*)d_ws;
  const size_t FEAT_ELTS = (size_t)NBATCH * NPTS * 128;   // 2,097,152
  float*    featA  = (float*)ws;                      ws += FEAT_ELTS * 4;            // 8 MB
  float*    featB  = (float*)ws;                      ws += FEAT_ELTS * 4;            // 8 MB
  float*    dist   = (float*)ws;                      ws += (size_t)NPTS * NPTS * 4;  // 16 MB
  unsigned* maxbuf = (unsigned*)ws;                   ws += FEAT_ELTS * 4;            // 8 MB
  int*      idxbuf = (int*)ws;                        ws += (size_t)NBATCH * NPTS * KK * 4;
  float*    sqbuf  = (float*)ws;                      ws += (size_t)NBATCH * NPTS * 4;
  float*    sums   = (float*)ws;                      ws += 256 * 4;
  float*    sumsq  = (float*)ws;                      ws += 256 * 4;

  // layer-1 input: transpose + pad (B,5,N) -> (B,N,8)
  {
    int total = NBATCH * NPTS * Cps[0];
    k_transpose_pad<<<(total + 255) / 256, 256, 0, stream>>>(x, featA, Cs[0], Cps[0]);
  }

  float* cur = featA;
  float* nxt = featB;
  for (int L = 0; L < 4; ++L) {
    const int C = Cs[L], Cp = Cps[L], O = Os[L];

    // point norms
    k_row_norms<<<(NBATCH * NPTS + 255) / 256, 256, 0, stream>>>(cur, sqbuf, Cp);

    // kNN: per-batch distance GEMM (WMMA f32) + top-k, reusing one NxN buffer
    for (int b = 0; b < NBATCH; ++b) {
      const int waves = (NPTS / 16) * (NPTS / 16);          // 16384
      k_dist_wmma<<<waves / 8, 256, 0, stream>>>(cur, sqbuf, dist, Cp, b);
      k_topk<<<NPTS, 256, 0, stream>>>(dist, idxbuf, b);
    }

    // reset stats + max accumulator
    {
      int total = NBATCH * NPTS * O;
      k_init_stats<<<(total + 255) / 256, 256, 0, stream>>>(sums, sumsq, maxbuf, O, total);
    }

    // fused edge-conv GEMM (WMMA f32) + BN stats + max-over-k
    {
      const int mtiles = (NBATCH * NPTS * KK) / 16;
      const int waves  = mtiles * (O / 16);
      k_edge_gemm_wmma<<<(waves * 32 + 255) / 256, 256, 0, stream>>>(
          cur, idxbuf, Wl[L], sums, sumsq, maxbuf, C, Cp, O);
    }

    // BN + LeakyReLU on max; last layer writes transposed straight to d_out
    {
      int total = NBATCH * NPTS * O;
      if (L < 3) {
        k_finalize<<<(total + 255) / 256, 256, 0, stream>>>(
            maxbuf, sums, sumsq, gl[L], bl[L], nxt, O, 0);
        float* tmp = cur; cur = nxt; nxt = tmp;
      } else {
        k_finalize<<<(total + 255) / 256, 256, 0, stream>>>(
            maxbuf, sums, sumsq, gl[L], bl[L], (float*)d_out, O, 1);
      }
    }
  }
}